// Refiner_71287867179398
// MI455X (gfx1250) — compile-verified
//
#include <hip/hip_runtime.h>
#include <math.h>
#include <cstdint>

#define KK 64
#define TSTEPS 10
#define BLOCK 128   // 4 waves; each wave gets a disjoint 16KB staging region

#ifndef M_PI_F
#define M_PI_F 3.14159265358979323846f
#endif

__device__ __forceinline__ float softplusf(float x) {
  return log1pf(__expf(x));
}

// Rotation-swizzled float4 index inside one array's per-wave region (512 float4s):
// element E (0..31), chunk q (0..15) -> E*16 + ((q+E)&15). Readback by lane L
// then hits bank group 4*((j+L)&15): only L and L+16 collide (2-way vs 32-way).
__device__ __forceinline__ int swz(int E, int q) {
  return (E << 4) | ((q + E) & 15);
}

__global__ __launch_bounds__(BLOCK) void fda_refine_kernel(
    const float* __restrict__ z_real, const float* __restrict__ z_imag,
    const float* __restrict__ theta0_deg, const float* __restrict__ r0_m,
    const float* __restrict__ alpha_raw, const float* __restrict__ lambda_raw,
    float* __restrict__ out_theta, float* __restrict__ out_r,
    float* __restrict__ out_J, int B)
{
  // 4 waves * 2 arrays * 512 float4 = 64KB LDS
  __shared__ float4 stage[4 * 2 * 512];

  const int tid  = threadIdx.x;
  const int lane = tid & 31;
  const int wave = tid >> 5;
  const int elem = blockIdx.x * BLOCK + tid;
  const int waveElemBase = blockIdx.x * BLOCK + wave * 32;

  float zr[KK], zi[KK];

  const bool fullWave = (waveElemBase + 32) <= B;  // uniform per wave

  if (fullWave) {
    float4* lr = &stage[(wave * 2 + 0) * 512];     // z_real region (this wave)
    float4* li = &stage[(wave * 2 + 1) * 512];     // z_imag region (this wave)
    const float4* gr = (const float4*)(z_real) + (size_t)waveElemBase * (KK / 4);
    const float4* gi = (const float4*)(z_imag) + (size_t)waveElemBase * (KK / 4);

    // Issue all 32 coalesced async global->LDS b128 copies, then one wait.
    #pragma unroll
    for (int j = 0; j < 16; ++j) {
      const int g = j * 32 + lane;                 // linear float4 idx in region
      const int E = g >> 4, q = g & 15;
      {
        unsigned dst = (unsigned)(uintptr_t)(lr + swz(E, q));
        unsigned long long src = (unsigned long long)(uintptr_t)(gr + g);
        asm volatile("global_load_async_to_lds_b128 %0, %1, off"
                     :: "v"(dst), "v"(src) : "memory");
      }
      {
        unsigned dst = (unsigned)(uintptr_t)(li + swz(E, q));
        unsigned long long src = (unsigned long long)(uintptr_t)(gi + g);
        asm volatile("global_load_async_to_lds_b128 %0, %1, off"
                     :: "v"(dst), "v"(src) : "memory");
      }
    }
    asm volatile("s_wait_asynccnt 0" ::: "memory");

    #pragma unroll
    for (int j = 0; j < 16; ++j) {
      float4 a = lr[swz(lane, j)];
      zr[4 * j + 0] = a.x; zr[4 * j + 1] = a.y;
      zr[4 * j + 2] = a.z; zr[4 * j + 3] = a.w;
      float4 b = li[swz(lane, j)];
      zi[4 * j + 0] = b.x; zi[4 * j + 1] = b.y;
      zi[4 * j + 2] = b.z; zi[4 * j + 3] = b.w;
    }
  } else if (elem < B) {
    // tail path (unused for B=262144, kept for safety): direct b128 loads
    const float4* gr = (const float4*)(z_real) + (size_t)elem * (KK / 4);
    const float4* gi = (const float4*)(z_imag) + (size_t)elem * (KK / 4);
    #pragma unroll
    for (int j = 0; j < 16; ++j) {
      float4 a = gr[j];
      zr[4 * j + 0] = a.x; zr[4 * j + 1] = a.y; zr[4 * j + 2] = a.z; zr[4 * j + 3] = a.w;
      float4 b = gi[j];
      zi[4 * j + 0] = b.x; zi[4 * j + 1] = b.y; zi[4 * j + 2] = b.z; zi[4 * j + 3] = b.w;
    }
  } else {
    #pragma unroll
    for (int k = 0; k < KK; ++k) { zr[k] = 0.f; zi[k] = 0.f; }
  }

  if (elem >= B) return;

  // ||z||^2 is iteration-invariant
  float szz = 0.f;
  #pragma unroll
  for (int k = 0; k < KK; ++k) szz += zr[k] * zr[k] + zi[k] * zi[k];

  // ---- initial u from (theta0, r0) via atanh / logit with clipping ----
  float th0 = theta0_deg[elem];
  float rr0 = r0_m[elem];
  float y = th0 * (1.f / 60.f);                         // (theta-TH_MID)/TH_HALF
  y = fminf(fmaxf(y, -1.f + 1e-6f), 1.f - 1e-6f);
  float u0 = 0.5f * (log1pf(y) - log1pf(-y));
  float pp = rr0 * (1.f / 2000.f);
  pp = fminf(fmaxf(pp, 1e-6f), 1.f - 1e-6f);
  float u1 = __logf(pp) - log1pf(-pp);

  const float TWO_PI = 6.283185307179586f;
  float theta = 0.f, r = 0.f, J = 0.f;

  #pragma unroll 1
  for (int t = 0; t <= TSTEPS; ++t) {
    // map u -> (theta, r)
    float e2 = __expf(2.f * u0);
    float tu = 1.f - 2.f / (e2 + 1.f);                  // tanh(u0)
    float sg = 1.f / (1.f + __expf(-u1));               // sigmoid(u1)
    theta = 60.f * tu;
    r = 2000.f * sg;
    float th = theta * (M_PI_F / 180.f);
    float sth, cth;
    __sincosf(th, &sth, &cth);
    float psi = 0.5f * sth - 2.0e-4f * r;               // F0*D/C=0.5, 2*DF/C=2e-4

    // w = exp(-i*2*pi*psi); 4 interleaved recurrence chains stepping w^4
    float ws, wc;
    __sincosf(-TWO_PI * psi, &ws, &wc);
    float w2r = wc * wc - ws * ws;
    float w2i = 2.f * wc * ws;
    float w4r = w2r * w2r - w2i * w2i;
    float w4i = 2.f * w2r * w2i;
    float er[4], ei[4];
    er[0] = 1.f;                 ei[0] = 0.f;
    er[1] = wc;                  ei[1] = ws;
    er[2] = w2r;                 ei[2] = w2i;
    er[3] = w2r * wc - w2i * ws; ei[3] = w2r * ws + w2i * wc;
    float a0r[4] = {0.f, 0.f, 0.f, 0.f}, a0i[4] = {0.f, 0.f, 0.f, 0.f};
    float a1r[4] = {0.f, 0.f, 0.f, 0.f}, a1i[4] = {0.f, 0.f, 0.f, 0.f};

    #pragma unroll
    for (int j = 0; j < 16; ++j) {
      #pragma unroll
      for (int c = 0; c < 4; ++c) {
        const int k = 4 * j + c;
        float tr = er[c] * zr[k] - ei[c] * zi[k];       // (e^{-i2pik psi} * z)_re
        float ti = er[c] * zi[k] + ei[c] * zr[k];       // (e^{-i2pik psi} * z)_im
        a0r[c] += tr; a0i[c] += ti;
        a1r[c] = fmaf((float)k, tr, a1r[c]);
        a1i[c] = fmaf((float)k, ti, a1i[c]);
        float nr = er[c] * w4r - ei[c] * w4i;           // advance chain by w^4
        float ni = er[c] * w4i + ei[c] * w4r;
        er[c] = nr; ei[c] = ni;
      }
    }
    float S0r = (a0r[0] + a0r[1]) + (a0r[2] + a0r[3]);
    float S0i = (a0i[0] + a0i[1]) + (a0i[2] + a0i[3]);
    float S1r = (a1r[0] + a1r[1]) + (a1r[2] + a1r[3]);
    float S1i = (a1i[0] + a1i[1]) + (a1i[2] + a1i[3]);

    float den = szz + 1e-12f;
    float invKden = 1.f / (64.f * den);
    J = (S0r * S0r + S0i * S0i) * invKden;
    if (t == TSTEPS) break;

    // analytic grad of -J wrt u
    float dJdpsi = (4.f * M_PI_F) * invKden * (S0r * S1i - S0i * S1r);
    float dpsi_du0 = 0.5f * cth * (M_PI_F / 3.f) * (1.f - tu * tu);
    float dpsi_du1 = -0.4f * sg * (1.f - sg);
    float g0 = -dJdpsi * dpsi_du0;
    float g1 = -dJdpsi * dpsi_du1;

    float al = softplusf(alpha_raw[t]) + 1e-8f;         // uniform -> s_load
    float la = softplusf(lambda_raw[t]) + 1e-8f;
    u0 -= al * g0 / (fabsf(g0) + la);
    u1 -= al * g1 / (fabsf(g1) + la);
  }

  out_theta[elem] = theta;
  out_r[elem]     = r;
  out_J[elem]     = J;
}

extern "C" void kernel_launch(void* const* d_in, const int* in_sizes, int n_in,
                              void* d_out, int out_size, void* d_ws, size_t ws_size,
                              hipStream_t stream) {
  const float* z_real     = (const float*)d_in[0];
  const float* z_imag     = (const float*)d_in[1];
  const float* theta0_deg = (const float*)d_in[2];
  const float* r0_m       = (const float*)d_in[3];
  const float* alpha_raw  = (const float*)d_in[4];
  const float* lambda_raw = (const float*)d_in[5];
  const int B = in_sizes[2];            // theta0 has B elements
  float* out = (float*)d_out;           // [theta_T | r_T | J_T], 3*B floats

  dim3 grid((B + BLOCK - 1) / BLOCK), block(BLOCK);
  hipLaunchKernelGGL(fda_refine_kernel, grid, block, 0, stream,
                     z_real, z_imag, theta0_deg, r0_m, alpha_raw, lambda_raw,
                     out, out + B, out + 2 * B, B);
}